// SCRF_41540923687006
// MI455X (gfx1250) — compile-verified
//
#include <hip/hip_runtime.h>

typedef float v2f __attribute__((ext_vector_type(2)));
typedef float v8f __attribute__((ext_vector_type(8)));

#define LC       768
#define DD       256
#define NT       12
#define START_ID 10
#define NEG_BIG  (-1.0e30f)

// ---------------------------------------------------------------------------
// WMMA f32 16x16x4 fragments.
// A-frag (ISA 7.12.2, 32-bit A 16x4): lane L (M = L&15):
//   lanes 0-15  -> v0=K(k+0), v1=K(k+1); lanes 16-31 -> v0=K(k+2), v1=K(k+3)
// B-frag (4x16, mirrored): lane L (N = L&15), same K-pair selection by lane-half.
// Both operands of x@W.T are row-major with K contiguous, so A and B fragments
// use the identical per-lane address pattern: row_ptr[k + 2*(lane>>4)].
// C/D layout: VGPR r -> row = r + 8*(lane>>4), col = lane&15.
// ---------------------------------------------------------------------------
__device__ __forceinline__ v8f wmma4(v2f a, v2f b, v8f c) {
  return __builtin_amdgcn_wmma_f32_16x16x4_f32(false, a, false, b,
                                               (short)0, c, false, false);
}

// 2x2 register-blocked K-loop: 4 b64 loads feed 4 WMMAs per k-step.
__device__ __forceinline__ void wmma_loop_2x2(const float* __restrict__ ap0,
                                              const float* __restrict__ ap1,
                                              const float* __restrict__ bp0,
                                              const float* __restrict__ bp1,
                                              int K,
                                              v8f& c00, v8f& c01, v8f& c10, v8f& c11) {
#pragma unroll 4
  for (int k = 0; k < K; k += 4) {
    v2f a0 = { ap0[k], ap0[k + 1] };
    v2f a1 = { ap1[k], ap1[k + 1] };
    v2f b0 = { bp0[k], bp0[k + 1] };
    v2f b1 = { bp1[k], bp1[k + 1] };
    c00 = wmma4(a0, b0, c00);
    c01 = wmma4(a0, b1, c01);
    c10 = wmma4(a1, b0, c10);
    c11 = wmma4(a1, b1, c11);
  }
}

// h = feats(768x512) @ Wd(256x512).T + bd  -> diag0 (768x256). 32x32 per wave.
__global__ void gemm_h(const float* __restrict__ A, const float* __restrict__ W,
                       const float* __restrict__ bias, float* __restrict__ C) {
  const int tm = blockIdx.x, tn = blockIdx.y;        // 24 x 8
  const int lane = threadIdx.x;
  const int half = lane >> 4, l16 = lane & 15;
  const float* ap0 = A + (size_t)(tm * 32 + l16) * 512 + 2 * half;
  const float* ap1 = ap0 + (size_t)16 * 512;
  const float* bp0 = W + (size_t)(tn * 32 + l16) * 512 + 2 * half;
  const float* bp1 = bp0 + (size_t)16 * 512;
  v8f c00 = {}, c01 = {}, c10 = {}, c11 = {};
  wmma_loop_2x2(ap0, ap1, bp0, bp1, 512, c00, c01, c10, c11);
  const float bn0 = bias[tn * 32 + l16];
  const float bn1 = bias[tn * 32 + 16 + l16];
#pragma unroll
  for (int r = 0; r < 8; ++r) {
    const size_t r0 = (size_t)(tm * 32 + r + 8 * half) * DD;
    const size_t r1 = (size_t)(tm * 32 + 16 + r + 8 * half) * DD;
    const int n0 = tn * 32 + l16;
    C[r0 + n0]      = c00[r] + bn0;
    C[r0 + n0 + 16] = c01[r] + bn1;
    C[r1 + n0]      = c10[r] + bn0;
    C[r1 + n0 + 16] = c11[r] + bn1;
  }
}

// P = ht(768x256) @ [Wl;Wr;Gl;Gr](2048x256).T -> P (768x2048). 32x32 per wave.
// 32-column spans never cross the 256/512/1280 weight boundaries.
__global__ void gemm_proj(const float* __restrict__ A,
                          const float* __restrict__ Wl, const float* __restrict__ Wr,
                          const float* __restrict__ Gl, const float* __restrict__ Gr,
                          float* __restrict__ P) {
  const int tm = blockIdx.x, tn = blockIdx.y;        // 24 x 64
  const int lane = threadIdx.x;
  const int half = lane >> 4, l16 = lane & 15;
  const int n0 = tn * 32;
  const float* W; int rb;
  if (n0 < 256)       { W = Wl; rb = n0; }
  else if (n0 < 512)  { W = Wr; rb = n0 - 256; }
  else if (n0 < 1280) { W = Gl; rb = n0 - 512; }
  else                { W = Gr; rb = n0 - 1280; }
  const float* ap0 = A + (size_t)(tm * 32 + l16) * DD + 2 * half;
  const float* ap1 = ap0 + (size_t)16 * DD;
  const float* bp0 = W + (size_t)(rb + l16) * DD + 2 * half;
  const float* bp1 = bp0 + (size_t)16 * DD;
  v8f c00 = {}, c01 = {}, c10 = {}, c11 = {};
  wmma_loop_2x2(ap0, ap1, bp0, bp1, DD, c00, c01, c10, c11);
#pragma unroll
  for (int r = 0; r < 8; ++r) {
    const size_t r0 = (size_t)(tm * 32 + r + 8 * half) * 2048;
    const size_t r1 = (size_t)(tm * 32 + 16 + r + 8 * half) * 2048;
    const int nn = n0 + l16;
    P[r0 + nn]      = c00[r];
    P[r0 + nn + 16] = c01[r];
    P[r1 + nn]      = c10[r];
    P[r1 + nn + 16] = c11[r];
  }
}

// Gating combine: ht_next[i] = softmax3(g)·{4σ(.)−2, l, r},  i in [0, Lm)
__global__ void combine_k(const float* __restrict__ P, const float* __restrict__ htp,
                          const float* __restrict__ bl, const float* __restrict__ br,
                          const float* __restrict__ bgl, const float* __restrict__ bgr,
                          float* __restrict__ htn, int Lm) {
  const int idx = blockIdx.x * 256 + threadIdx.x;
  if (idx >= Lm * DD) return;
  const int i = idx >> 8, d = idx & 255;
  const float* Pi = P + (size_t)i * 2048;
  const float* Pj = Pi + 2048;
  const float a  = Pi[d] + bl[d] + Pj[256 + d] + br[d];
  const float hh = 4.0f / (1.0f + expf(-a)) - 2.0f;
  const float g0 = Pi[512 + d]  + bgl[d]       + Pj[1280 + d] + bgr[d];
  const float g1 = Pi[768 + d]  + bgl[256 + d] + Pj[1536 + d] + bgr[256 + d];
  const float g2 = Pi[1024 + d] + bgl[512 + d] + Pj[1792 + d] + bgr[512 + d];
  const float mg = fmaxf(g0, fmaxf(g1, g2));
  const float e0 = expf(g0 - mg), e1 = expf(g1 - mg), e2 = expf(g2 - mg);
  const float inv = 1.0f / (e0 + e1 + e2);
  const float l = htp[(size_t)i * DD + d], r = htp[(size_t)(i + 1) * DD + d];
  htn[(size_t)i * DD + d] = (e0 * hh + e1 * l + e2 * r) * inv;
}

// tagsc[sl*768+i][c] = diag[sl][i]·Wt[c] + bt[c]. 2x1 M-blocked, B reused.
__global__ void gemm_tag(const float* __restrict__ diag, const float* __restrict__ Wt,
                         const float* __restrict__ bt, float* __restrict__ tagsc) {
  const int tm = blockIdx.x;                         // 192 blocks over M = 8*768
  const int lane = threadIdx.x;
  const int half = lane >> 4, l16 = lane & 15;
  const int rn = (l16 < NT) ? l16 : (NT - 1);        // clamp Wt row (avoid OOB)
  const float* ap0 = diag + (size_t)(tm * 32 + l16) * DD + 2 * half;
  const float* ap1 = ap0 + (size_t)16 * DD;
  const float* bp = Wt + (size_t)rn * DD + 2 * half;
  v8f c0 = {}, c1 = {};
#pragma unroll 4
  for (int k = 0; k < DD; k += 4) {
    v2f a0 = { ap0[k], ap0[k + 1] };
    v2f a1 = { ap1[k], ap1[k + 1] };
    v2f b  = { bp[k],  bp[k + 1] };
    c0 = wmma4(a0, b, c0);
    c1 = wmma4(a1, b, c1);
  }
  const float bv = (l16 < NT) ? bt[l16] : 0.0f;
#pragma unroll
  for (int r = 0; r < 8; ++r) {
    tagsc[(size_t)(tm * 32 + r + 8 * half) * 16 + l16]      = c0[r] + bv;
    tagsc[(size_t)(tm * 32 + 16 + r + 8 * half) * 16 + l16] = c1[r] + bv;
  }
}

// ---------------------------------------------------------------------------
// Gold gather + sequential semi-CRF forward scan. Single block.
// tid = p*12 + c over 12x12 (prev,cur) tags; 8-slot alpha ring in LDS.
// ---------------------------------------------------------------------------
__global__ void fwd_kernel(const float* __restrict__ tagsc, const float* __restrict__ trans,
                           const float* __restrict__ bt, const int* __restrict__ tags,
                           float* __restrict__ out) {
  __shared__ float s_trans[NT * NT];
  __shared__ float ring[8 * NT];     // ring[(alpha_idx & 7)*12 + tag]
  __shared__ float redm[NT * NT];
  __shared__ float reds[NT * NT];
  __shared__ float s_gold;
  const int tid = threadIdx.x;

  if (tid < NT * NT) s_trans[tid] = trans[tid];
  if (tid < 8 * NT)  ring[tid] = -10000.0f;
  if (tid == 0)      ring[START_ID] = 0.0f;            // slot 0 = alpha_0
  __syncthreads();

  // gold score: 96 spans, indices i,j,prev,cur all in [0,12)
  if (tid < 96) {
    const int gi = tags[tid * 4 + 0], gj = tags[tid * 4 + 1];
    const int gp = tags[tid * 4 + 2], gc = tags[tid * 4 + 3];
    const int dlt = gj - gi;
    const float base = (dlt >= 0 && dlt < 8)
                         ? tagsc[(size_t)(dlt * LC + gi) * 16 + gc]   // incl. bt
                         : bt[gc];                                    // zero-score region
    redm[tid] = base + s_trans[gp * NT + gc];
  }
  __syncthreads();
  if (tid == 0) {
    float g = 0.0f;
    for (int s = 0; s < 96; ++s) g += redm[s];
    s_gold = g;
  }
  __syncthreads();

  const int p = tid / NT, c = tid % NT;
  const float tr = (tid < NT * NT) ? s_trans[tid] : 0.0f;

  for (int i = 1; i <= LC; ++i) {
    float m = NEG_BIG, s = 0.0f;
    if (tid < NT * NT) {
#pragma unroll
      for (int k = 0; k < 8; ++k) {
        const int aidx = i - 8 + k;                  // span start position
        if (aidx >= 0) {
          const int sl = 7 - k;                      // span length - 1
          const float t = tagsc[(size_t)(sl * LC + aidx) * 16 + c] + tr
                        + ring[(aidx & 7) * NT + p];
          if (t > m) { s = s * expf(m - t) + 1.0f; m = t; }
          else       { s += expf(t - m); }
        }
      }
      redm[tid] = m; reds[tid] = s;
    }
    __syncthreads();
    if (tid < NT) {                                  // reduce over prev tag p
      float M = NEG_BIG, S = 0.0f;
      for (int q = 0; q < NT; ++q) {
        const float mq = redm[q * NT + tid], sq = reds[q * NT + tid];
        if (mq > M) { S = S * expf(M - mq) + sq; M = mq; }
        else        { S += sq * expf(mq - M); }
      }
      ring[(i & 7) * NT + tid] = M + logf(S);        // alpha_i[cur]
    }
    __syncthreads();
  }

  if (tid == 0) {
    float lz = 0.0f;                                 // alphas[-1, :STOP_ID].sum()
    for (int cc = 0; cc < 11; ++cc) lz += ring[((LC) & 7) * NT + cc];
    out[0] = lz - s_gold;
  }
}

// ---------------------------------------------------------------------------
extern "C" void kernel_launch(void* const* d_in, const int* in_sizes, int n_in,
                              void* d_out, int out_size, void* d_ws, size_t ws_size,
                              hipStream_t stream) {
  (void)in_sizes; (void)n_in; (void)out_size; (void)ws_size;
  const float* feats = (const float*)d_in[0];
  const int*   tags  = (const int*)  d_in[1];
  const float* Wd    = (const float*)d_in[2];
  const float* bd    = (const float*)d_in[3];
  const float* Wl    = (const float*)d_in[4];
  const float* bl    = (const float*)d_in[5];
  const float* Wr    = (const float*)d_in[6];
  const float* br    = (const float*)d_in[7];
  const float* Gl    = (const float*)d_in[8];
  const float* bgl   = (const float*)d_in[9];
  const float* Gr    = (const float*)d_in[10];
  const float* bgr   = (const float*)d_in[11];
  const float* Wt    = (const float*)d_in[12];
  const float* bt    = (const float*)d_in[13];
  const float* trans = (const float*)d_in[14];
  float* out = (float*)d_out;

  float* ws    = (float*)d_ws;
  float* diag  = ws;                               // 8*768*256
  float* P     = diag + (size_t)8 * LC * DD;       // 768*2048
  float* tagsc = P + (size_t)LC * 2048;            // 8*768*16

  // h -> diag[0]
  { dim3 g(LC / 32, DD / 32);
    gemm_h<<<g, 32, 0, stream>>>(feats, Wd, bd, diag); }

  // sequential span-composition recurrence
  for (int sl = 1; sl <= 7; ++sl) {
    const float* htp = diag + (size_t)(sl - 1) * LC * DD;
    float* htn = diag + (size_t)sl * LC * DD;
    dim3 g(LC / 32, 2048 / 32);
    gemm_proj<<<g, 32, 0, stream>>>(htp, Wl, Wr, Gl, Gr, P);
    const int Lm = LC - sl;
    combine_k<<<(Lm * DD + 255) / 256, 256, 0, stream>>>(P, htp, bl, br, bgl, bgr, htn, Lm);
  }

  // tag scores for all diagonals
  { dim3 g((8 * LC) / 32, 1);
    gemm_tag<<<g, 32, 0, stream>>>(diag, Wt, bt, tagsc); }

  // gold + forward scan -> scalar
  fwd_kernel<<<1, 160, 0, stream>>>(tagsc, trans, bt, tags, out);
}